// MoERouter_42047729827841
// MI455X (gfx1250) — compile-verified
//
#include <hip/hip_runtime.h>
#include <hip/hip_bf16.h>

typedef __attribute__((ext_vector_type(2))) float v2f;
typedef __attribute__((ext_vector_type(4))) float v4f;
typedef __attribute__((ext_vector_type(8))) float v8f;

#define D_DIM 2048
#define E_NUM 8
#define TOPK 2
#define WAVES_PER_BLOCK 4
#define TOK_PER_WAVE 16
#define TOK_PER_BLOCK (WAVES_PER_BLOCK * TOK_PER_WAVE)
#define LDS_STRIDE 9   // 8 experts + 1 pad word -> conflict-free banks
#define CHUNK 32       // K-elements per pipeline stage (8 WMMAs)

// Issue the 16 loads for one 32-wide K chunk into a register buffer.
#define LOAD_CHUNK(kbase, areg, breg)                                   \
    do {                                                                \
        _Pragma("unroll")                                               \
        for (int kk = 0; kk < 8; ++kk) {                                \
            areg[kk] = *(const v2f*)(arow + (kbase) + kk * 4);          \
            breg[kk] = *(const v2f*)(brow + (kbase) + kk * 4);          \
        }                                                               \
    } while (0)

// Consume one staged chunk: 8 WMMAs alternating accumulators.
#define COMPUTE_CHUNK(areg, breg)                                       \
    do {                                                                \
        _Pragma("unroll")                                               \
        for (int kk = 0; kk < 8; kk += 2) {                             \
            v2f b0; b0.x = breg[kk].x * bscale;                         \
            b0.y = breg[kk].y * bscale;                                 \
            acc0 = __builtin_amdgcn_wmma_f32_16x16x4_f32(               \
                false, areg[kk], false, b0, (short)0, acc0,             \
                false, false);                                          \
            v2f b1; b1.x = breg[kk + 1].x * bscale;                     \
            b1.y = breg[kk + 1].y * bscale;                             \
            acc1 = __builtin_amdgcn_wmma_f32_16x16x4_f32(               \
                false, areg[kk + 1], false, b1, (short)0, acc1,         \
                false, false);                                          \
        }                                                               \
    } while (0)

__global__ __launch_bounds__(WAVES_PER_BLOCK * 32)
void moe_router_wmma(const float* __restrict__ x,
                     const float* __restrict__ gw,
                     float* __restrict__ out_w,
                     float* __restrict__ out_mask,
                     float* __restrict__ out_lg,
                     int N)
{
    __shared__ float lds[WAVES_PER_BLOCK * TOK_PER_WAVE * LDS_STRIDE];

    const int lane = threadIdx.x & 31;
    const int wave = threadIdx.x >> 5;
    const int sub  = lane & 15;   // token row (A) / expert column (B)
    const int half = lane >> 4;   // K sub-offset selector

    const int t0 = (blockIdx.x * WAVES_PER_BLOCK + wave) * TOK_PER_WAVE;
    if (t0 >= N) return;  // wave-uniform

    // A fragment: lane holds 2 consecutive f32 of its token row (16x4 layout).
    const float* __restrict__ arow = x + (size_t)(t0 + sub) * D_DIM + half * 2;
    // B fragment: expert row (clamped); experts 8..15 zeroed via bscale.
    const float* __restrict__ brow = gw + (size_t)(sub & (E_NUM - 1)) * D_DIM + half * 2;
    const float bscale = (sub < E_NUM) ? 1.0f : 0.0f;

    v8f acc0 = {}, acc1 = {};
    v2f a0[8], b0buf[8], a1[8], b1buf[8];

    // Software pipeline: stage chunk k+CHUNK while computing chunk k.
    LOAD_CHUNK(0, a0, b0buf);
    for (int k = 0; k < D_DIM; k += 2 * CHUNK) {
        __builtin_prefetch(arow + k + 512, 0, 0);   // ~2KB ahead into L2
        LOAD_CHUNK(k + CHUNK, a1, b1buf);
        COMPUTE_CHUNK(a0, b0buf);
        if (k + 2 * CHUNK < D_DIM) {
            LOAD_CHUNK(k + 2 * CHUNK, a0, b0buf);
        }
        COMPUTE_CHUNK(a1, b1buf);
    }
    v8f acc = acc0 + acc1;

    // Scatter logits to LDS: lane (e,half) holds tokens half*8..half*8+7 of expert e.
    float* my = lds + wave * (TOK_PER_WAVE * LDS_STRIDE);
    if (sub < E_NUM) {
        #pragma unroll
        for (int j = 0; j < 8; ++j)
            my[(half * 8 + j) * LDS_STRIDE + sub] = acc[j];
    }
    __syncthreads();

    // Per-token epilogue: lanes 0..15 each own one token.
    if (half == 0) {
        const int n = t0 + sub;

        float lg[E_NUM];
        #pragma unroll
        for (int e = 0; e < E_NUM; ++e) lg[e] = my[sub * LDS_STRIDE + e];

        // softmax (fp32)
        float mx = lg[0];
        #pragma unroll
        for (int e = 1; e < E_NUM; ++e) mx = fmaxf(mx, lg[e]);
        float p[E_NUM];
        float sum = 0.0f;
        #pragma unroll
        for (int e = 0; e < E_NUM; ++e) { p[e] = __expf(lg[e] - mx); sum += p[e]; }
        const float inv_sum = 1.0f / sum;
        #pragma unroll
        for (int e = 0; e < E_NUM; ++e) p[e] *= inv_sum;

        // top-2, first occurrence wins on ties (matches jax.lax.top_k)
        int i0 = 0; float p0 = p[0];
        #pragma unroll
        for (int e = 1; e < E_NUM; ++e) if (p[e] > p0) { p0 = p[e]; i0 = e; }
        int i1 = -1; float p1 = -__builtin_inff();
        #pragma unroll
        for (int e = 0; e < E_NUM; ++e) if (e != i0 && p[e] > p1) { p1 = p[e]; i1 = e; }

        // normalized top-2 weights
        const float winv = 1.0f / (p0 + p1);
        v2f w2; w2.x = p0 * winv; w2.y = p1 * winv;
        *(v2f*)(out_w + (size_t)TOPK * n) = w2;

        // one-hot expert mask [E, K, N] as 0.0/1.0
        #pragma unroll
        for (int e = 0; e < E_NUM; ++e) {
            out_mask[((size_t)e * TOPK + 0) * (size_t)N + n] = (e == i0) ? 1.0f : 0.0f;
            out_mask[((size_t)e * TOPK + 1) * (size_t)N + n] = (e == i1) ? 1.0f : 0.0f;
        }

        // raw gate logits [N, E]
        v4f lo; lo.x = lg[0]; lo.y = lg[1]; lo.z = lg[2]; lo.w = lg[3];
        v4f hi; hi.x = lg[4]; hi.y = lg[5]; hi.z = lg[6]; hi.w = lg[7];
        *(v4f*)(out_lg + (size_t)n * E_NUM)     = lo;
        *(v4f*)(out_lg + (size_t)n * E_NUM + 4) = hi;
    }
}

extern "C" void kernel_launch(void* const* d_in, const int* in_sizes, int n_in,
                              void* d_out, int out_size, void* d_ws, size_t ws_size,
                              hipStream_t stream) {
    const float* x  = (const float*)d_in[0];
    const float* gw = (const float*)d_in[1];
    const int N = in_sizes[0] / D_DIM;  // 4*4096 = 16384 tokens

    float* out      = (float*)d_out;
    float* out_w    = out;                                        // [N, 2]
    float* out_mask = out + (size_t)N * TOPK;                     // [8, 2, N]
    float* out_lg   = out + (size_t)N * TOPK
                          + (size_t)E_NUM * TOPK * (size_t)N;     // [N, 8]

    const int blocks = (N + TOK_PER_BLOCK - 1) / TOK_PER_BLOCK;   // 256
    moe_router_wmma<<<blocks, WAVES_PER_BLOCK * 32, 0, stream>>>(
        x, gw, out_w, out_mask, out_lg, N);
}